// SchNetInteraction_44332652429577
// MI455X (gfx1250) — compile-verified
//
#include <hip/hip_runtime.h>

#define BATCH 16
#define NATOM 256
#define NBH   64
#define NF    128
#define NS    50

typedef float v2f __attribute__((ext_vector_type(2)));
typedef float v8f __attribute__((ext_vector_type(8)));

// Full-precision CDNA5 WMMA: D(16x16,f32) = A(16x4,f32) * B(4x16,f32) + C
__device__ __forceinline__ v8f wmma4(v2f a, v2f b, v8f c) {
    return __builtin_amdgcn_wmma_f32_16x16x4_f32(false, a, false, b, (short)0, c, false, false);
}

__device__ __forceinline__ v8f splat8(float v) { v8f r = {v, v, v, v, v, v, v, v}; return r; }

// shifted softplus: softplus(x) - ln2, numerically stable
__device__ __forceinline__ float sspf(float x) {
    return fmaxf(x, 0.0f) + log1pf(__expf(-fabsf(x))) - 0.69314718055994530942f;
}

// Wave-level GEMM: 16x128 C-tile, K=128. A rows [rowbase,rowbase+16) in LDS
// (row stride lda, padded to avoid bank conflicts), B row-major (128 x NF) in
// global (weights: L2-resident, broadcast-friendly across waves/blocks).
__device__ __forceinline__ void gemm_k128(const float* As, int lda, int rowbase,
                                          const float* __restrict__ Bg,
                                          v8f acc[8], int lane) {
    const int half = lane >> 4, lm = lane & 15;
    const float* arow = As + (rowbase + lm) * lda;
    for (int k = 0; k < NF; k += 4) {
        const int krow = k + 2 * half;
        v2f a; a.x = arow[krow]; a.y = arow[krow + 1];
        const float* B0 = Bg + krow * NF;
        #pragma unroll
        for (int t = 0; t < 8; ++t) {
            v2f b; b.x = B0[t * 16 + lm]; b.y = B0[NF + t * 16 + lm];
            acc[t] = wmma4(a, b, acc[t]);
        }
    }
}

// ---------------- kernel 1: y = x @ w_in2f (no bias) ----------------
__global__ __launch_bounds__(128) void k_in2f(const float* __restrict__ x,
                                              const float* __restrict__ w,
                                              float* __restrict__ y) {
    __shared__ float sA[64 * 132];
    const int tid = threadIdx.x, lane = tid & 31, wave = tid >> 5;
    const int half = lane >> 4, lm = lane & 15;
    const size_t base = (size_t)blockIdx.x * 64 * NF;
    for (int i = tid; i < 64 * NF; i += 128)
        sA[(i >> 7) * 132 + (i & 127)] = x[base + i];
    __syncthreads();
    const int rowbase = wave * 16;
    v8f acc[8];
    #pragma unroll
    for (int t = 0; t < 8; ++t) acc[t] = splat8(0.0f);
    gemm_k128(sA, 132, rowbase, w, acc, lane);
    #pragma unroll
    for (int t = 0; t < 8; ++t)
        #pragma unroll
        for (int r = 0; r < 8; ++r) {
            int row = rowbase + r + 8 * half;
            y[base + (size_t)row * NF + t * 16 + lm] = acc[t][r];
        }
}

// ---- kernel 2: fused filter-net + cutoff + gather + cfconv aggregation ----
// One block per (batch, atom). W-tile (64x128) never touches HBM.
__global__ __launch_bounds__(128) void k_cfconv(
    const float* __restrict__ y,      // (B*A, F) from k_in2f
    const float* __restrict__ r_ij,   // (B,A,NBH)
    const int*   __restrict__ nbrs,   // (B,A,NBH)
    const int*   __restrict__ nmask,  // (B,A,NBH)
    const float* __restrict__ f_ij,   // (B,A,NBH,S)
    const float* __restrict__ w1, const float* __restrict__ b1,
    const float* __restrict__ w2, const float* __restrict__ b2,
    float* __restrict__ agg)          // (B*A, F)
{
    __shared__ float sF[NBH * 52];    // f_ij tile, K padded 50->52 with zeros
    __shared__ float sH[NBH * 132];   // ssp(F@W1+b1)
    __shared__ float sY[NBH * 132];   // gathered neighbor feats * cutoff * mask
    __shared__ float sCoef[NBH];
    __shared__ int   sNbr[NBH];
    __shared__ float sAgg[NF];
    const int tid = threadIdx.x, lane = tid & 31, wave = tid >> 5;
    const int half = lane >> 4, lm = lane & 15;
    const int ba = blockIdx.x;        // 0..B*A-1
    const int bb = ba >> 8;           // batch index (A=256)
    const float* fij = f_ij + (size_t)ba * NBH * NS;
    for (int i = tid; i < NBH * 52; i += 128) {
        int n = i / 52, s = i - n * 52;
        sF[i] = (s < NS) ? fij[n * NS + s] : 0.0f;
    }
    if (tid < NBH) {
        float r = r_ij[(size_t)ba * NBH + tid];
        float c = 0.5f * (__cosf(r * 0.62831853071795864769f) + 1.0f); // pi/CUTOFF
        c = (r < 5.0f) ? c : 0.0f;
        c *= (float)nmask[(size_t)ba * NBH + tid];
        sCoef[tid] = c;
        sNbr[tid]  = nbrs[(size_t)ba * NBH + tid];
    }
    sAgg[tid] = 0.0f;
    __syncthreads();
    // gather neighbor features, pre-scaled by cutoff*mask (coalesced 512B rows)
    for (int n = 0; n < NBH; ++n)
        sY[n * 132 + tid] = y[((size_t)bb * NATOM + sNbr[n]) * NF + tid] * sCoef[n];

    // GEMM1: H = ssp(F(64x50) @ W1(50x128) + b1), K padded to 52
    const int rowbase = wave * 16;
    v8f acc[8];
    #pragma unroll
    for (int t = 0; t < 8; ++t) acc[t] = splat8(b1[t * 16 + lm]);
    const float* arow = sF + (rowbase + lm) * 52;
    for (int k = 0; k < 52; k += 4) {
        const int krow = k + 2 * half;
        v2f a; a.x = arow[krow]; a.y = arow[krow + 1];   // pad cols are zero
        const bool v0 = (krow < NS), v1 = (krow + 1 < NS);
        const float* B0 = w1 + krow * NF;
        #pragma unroll
        for (int t = 0; t < 8; ++t) {
            v2f b;
            b.x = v0 ? B0[t * 16 + lm] : 0.0f;
            b.y = v1 ? B0[NF + t * 16 + lm] : 0.0f;
            acc[t] = wmma4(a, b, acc[t]);
        }
    }
    #pragma unroll
    for (int t = 0; t < 8; ++t)
        #pragma unroll
        for (int r = 0; r < 8; ++r) {
            int row = rowbase + r + 8 * half;
            sH[row * 132 + t * 16 + lm] = sspf(acc[t][r]);
        }
    __syncthreads();
    // GEMM2: Wfilter = H @ W2 + b2; fuse cutoff/mask/gather product + row-sum
    #pragma unroll
    for (int t = 0; t < 8; ++t) acc[t] = splat8(b2[t * 16 + lm]);
    gemm_k128(sH, 132, rowbase, w2, acc, lane);
    #pragma unroll
    for (int t = 0; t < 8; ++t) {
        float p = 0.0f;
        #pragma unroll
        for (int r = 0; r < 8; ++r) {
            int row = rowbase + r + 8 * half;
            p += acc[t][r] * sY[row * 132 + t * 16 + lm];
        }
        atomicAdd(&sAgg[t * 16 + lm], p);   // ds_add_f32, 8 adders per column
    }
    __syncthreads();
    agg[(size_t)ba * NF + tid] = sAgg[tid];
}

// ---- kernel 3: out = ssp(agg @ w_f2out + b) @ w_dense + b ----
__global__ __launch_bounds__(128) void k_out(
    const float* __restrict__ agg,
    const float* __restrict__ w3, const float* __restrict__ b3,
    const float* __restrict__ w4, const float* __restrict__ b4,
    float* __restrict__ out)
{
    __shared__ float sA[64 * 132];
    __shared__ float sV[64 * 132];
    const int tid = threadIdx.x, lane = tid & 31, wave = tid >> 5;
    const int half = lane >> 4, lm = lane & 15;
    const size_t base = (size_t)blockIdx.x * 64 * NF;
    for (int i = tid; i < 64 * NF; i += 128)
        sA[(i >> 7) * 132 + (i & 127)] = agg[base + i];
    __syncthreads();
    const int rowbase = wave * 16;
    v8f acc[8];
    #pragma unroll
    for (int t = 0; t < 8; ++t) acc[t] = splat8(b3[t * 16 + lm]);
    gemm_k128(sA, 132, rowbase, w3, acc, lane);
    #pragma unroll
    for (int t = 0; t < 8; ++t)
        #pragma unroll
        for (int r = 0; r < 8; ++r) {
            int row = rowbase + r + 8 * half;
            sV[row * 132 + t * 16 + lm] = sspf(acc[t][r]);
        }
    __syncthreads();
    #pragma unroll
    for (int t = 0; t < 8; ++t) acc[t] = splat8(b4[t * 16 + lm]);
    gemm_k128(sV, 132, rowbase, w4, acc, lane);
    #pragma unroll
    for (int t = 0; t < 8; ++t)
        #pragma unroll
        for (int r = 0; r < 8; ++r) {
            int row = rowbase + r + 8 * half;
            out[base + (size_t)row * NF + t * 16 + lm] = acc[t][r];
        }
}

extern "C" void kernel_launch(void* const* d_in, const int* in_sizes, int n_in,
                              void* d_out, int out_size, void* d_ws, size_t ws_size,
                              hipStream_t stream) {
    const float* x        = (const float*)d_in[0];
    const float* r_ij     = (const float*)d_in[1];
    const int*   nbrs     = (const int*)  d_in[2];
    const int*   nmask    = (const int*)  d_in[3];
    const float* f_ij     = (const float*)d_in[4];
    const float* w_filt1  = (const float*)d_in[5];
    const float* b_filt1  = (const float*)d_in[6];
    const float* w_filt2  = (const float*)d_in[7];
    const float* b_filt2  = (const float*)d_in[8];
    const float* w_in2f   = (const float*)d_in[9];
    const float* w_f2out  = (const float*)d_in[10];
    const float* b_f2out  = (const float*)d_in[11];
    const float* w_dense  = (const float*)d_in[12];
    const float* b_dense  = (const float*)d_in[13];
    float* out = (float*)d_out;

    const int rowsTotal = BATCH * NATOM;          // 4096
    float* y   = (float*)d_ws;                    // 4096*128 f32 = 2 MB
    float* agg = y + (size_t)rowsTotal * NF;      // 2 MB more

    k_in2f  <<<rowsTotal / 64, 128, 0, stream>>>(x, w_in2f, y);
    k_cfconv<<<rowsTotal,      128, 0, stream>>>(y, r_ij, nbrs, nmask, f_ij,
                                                 w_filt1, b_filt1, w_filt2, b_filt2, agg);
    k_out   <<<rowsTotal / 64, 128, 0, stream>>>(agg, w_f2out, b_f2out, w_dense, b_dense, out);
}